// Net_int_2Edges_20272245637595
// MI455X (gfx1250) — compile-verified
//
#include <hip/hip_runtime.h>
#include <hip/hip_bf16.h>

typedef __attribute__((ext_vector_type(2))) float v2f;
typedef __attribute__((ext_vector_type(8))) float v8f;

#define SLOPE 0.22916666666666666f   // (1/8 + 1/3)/2
#define BN_EPS 1e-5f

static __device__ __forceinline__ float rrelu_f(float x) {
    return x >= 0.0f ? x : x * SLOPE;
}
static __device__ __forceinline__ float sigmoid_f(float x) {
    return 1.0f / (1.0f + __expf(-x));
}
static __device__ __forceinline__ v8f wmma4(v2f a, v2f b, v8f c) {
    return __builtin_amdgcn_wmma_f32_16x16x4_f32(false, a, false, b, (short)0, c, false, false);
}

// ---------------------------------------------------------------------------
// Generic pointwise utilities
// ---------------------------------------------------------------------------
__global__ void zero_kernel(float* p, int n) {
    int i = blockIdx.x * blockDim.x + threadIdx.x;
    if (i < n) p[i] = 0.0f;
}

// y[n,c] = (x[n,c]-m[c])*rsqrt(v[c]+eps)*g[c] + be[c]
__global__ void bn_kernel(const float* __restrict__ x, float* __restrict__ y,
                          const float* __restrict__ g, const float* __restrict__ be,
                          const float* __restrict__ m, const float* __restrict__ v,
                          int total, int C) {
    int i = blockIdx.x * blockDim.x + threadIdx.x;
    if (i >= total) return;
    int c = i % C;
    y[i] = (x[i] - m[c]) * rsqrtf(v[c] + BN_EPS) * g[c] + be[c];
}

// ---------------------------------------------------------------------------
// Register-blocked WMMA fp32 GEMM:
//   C[M,Nc] = act( A[M,K] @ W^T + bias ),  W is [Nc,K] row-major.
// One wave computes a 32x64 macro-tile (2 M-tiles x 4 N-tiles, 8 accumulators)
// with V_WMMA_F32_16X16X4_F32, K stepped by 4.
// Per K-step: 2 A-frag + 4 B-frag b64 loads feed 8 WMMAs (0.75 loads/WMMA).
// Requires M % 32 == 0, Nc % 64 == 0, K % 4 == 0.
// ---------------------------------------------------------------------------
__global__ void wmma_gemm_kernel(const float* __restrict__ A,
                                 const float* __restrict__ W,
                                 const float* __restrict__ bias,
                                 float* __restrict__ C,
                                 int M, int Nc, int K, int act, int totalGroups) {
    int wavesPerBlock = blockDim.x >> 5;
    int wave = threadIdx.x >> 5;
    int lane = threadIdx.x & 31;
    int grp = blockIdx.x * wavesPerBlock + wave;
    if (grp >= totalGroups) return;              // wave-uniform: EXEC stays all-1s

    int ngrpN = Nc >> 6;                         // groups of 64 output cols
    int gm = grp / ngrpN;
    int gn = grp - gm * ngrpN;

    int m16 = lane & 15;           // row (A) / col (B) sub-index
    int h   = lane >> 4;           // half select: VGPR0 holds K0/K2, VGPR1 K1/K3

    const float* arow0 = A + (size_t)(gm * 32 + m16) * K;
    const float* arow1 = arow0 + (size_t)16 * K;
    const float* wrow  = W + (size_t)(gn * 64 + m16) * K;

    v8f acc[2][4] = {};
    for (int k0 = 0; k0 < K; k0 += 4) {
        int ko = k0 + 2 * h;
        v2f a0 = *(const v2f*)(arow0 + ko);
        v2f a1 = *(const v2f*)(arow1 + ko);
        v2f b0 = *(const v2f*)(wrow + ko);
        v2f b1 = *(const v2f*)(wrow + (size_t)16 * K + ko);
        v2f b2 = *(const v2f*)(wrow + (size_t)32 * K + ko);
        v2f b3 = *(const v2f*)(wrow + (size_t)48 * K + ko);
        acc[0][0] = wmma4(a0, b0, acc[0][0]);
        acc[0][1] = wmma4(a0, b1, acc[0][1]);
        acc[0][2] = wmma4(a0, b2, acc[0][2]);
        acc[0][3] = wmma4(a0, b3, acc[0][3]);
        acc[1][0] = wmma4(a1, b0, acc[1][0]);
        acc[1][1] = wmma4(a1, b1, acc[1][1]);
        acc[1][2] = wmma4(a1, b2, acc[1][2]);
        acc[1][3] = wmma4(a1, b3, acc[1][3]);
    }

    #pragma unroll
    for (int j = 0; j < 4; ++j) {
        int ncol = gn * 64 + j * 16 + m16;
        float bv = bias ? bias[ncol] : 0.0f;
        #pragma unroll
        for (int i = 0; i < 2; ++i) {
            #pragma unroll
            for (int v = 0; v < 8; ++v) {
                int mrow = gm * 32 + i * 16 + v + 8 * h;  // lanes 16-31 hold M+8
                float val = acc[i][j][v] + bv;
                if (act) val = rrelu_f(val);
                C[(size_t)mrow * Nc + ncol] = val;
            }
        }
    }
}

// ---------------------------------------------------------------------------
// Edge embedding: ea[E,12] = rrelu(edge_attr[E,19] @ le_W^T + le_b)   (K=19)
// ---------------------------------------------------------------------------
__global__ void edge_embed_kernel(const float* __restrict__ attr,
                                  const float* __restrict__ Wl,
                                  const float* __restrict__ bl,
                                  float* __restrict__ ea, int E) {
    int i = blockIdx.x * blockDim.x + threadIdx.x;
    if (i >= E * 12) return;
    int e = i / 12, c = i - e * 12;
    const float* a = attr + (size_t)e * 19;
    const float* w = Wl + (size_t)c * 19;
    float s = bl[c];
    #pragma unroll
    for (int k = 0; k < 19; ++k) s += a[k] * w[k];
    ea[i] = rrelu_f(s);
}

// ---------------------------------------------------------------------------
// Weight repacks so the Z GEMMs read contiguous K:
//   Wcat1[c][i] = nn1_W[(i*64 + c%64)*12 + c/64]   (c<768, i<64)
//   Wcat2[c][i] = nn2_W[(i*128 + c%128)*8 + c/128] (c<1024, i<128)
// ---------------------------------------------------------------------------
__global__ void repack_w1_kernel(const float* __restrict__ nn1, float* __restrict__ Wc) {
    int idx = blockIdx.x * blockDim.x + threadIdx.x;
    if (idx >= 768 * 64) return;
    int c = idx >> 6, i = idx & 63;
    Wc[c * 64 + i] = nn1[(size_t)(i * 64 + (c & 63)) * 12 + (c >> 6)];
}
__global__ void repack_w2_kernel(const float* __restrict__ nn2, float* __restrict__ Wc) {
    int idx = blockIdx.x * blockDim.x + threadIdx.x;
    if (idx >= 1024 * 128) return;
    int c = idx >> 7, i = idx & 127;
    Wc[c * 128 + i] = nn2[(size_t)(i * 128 + (c & 127)) * 8 + (c >> 7)];
}

// ---------------------------------------------------------------------------
// Degrees
// ---------------------------------------------------------------------------
__global__ void deg1_kernel(const int* __restrict__ tgt, float* __restrict__ deg, int E) {
    int e = blockIdx.x * blockDim.x + threadIdx.x;
    if (e < E) atomicAdd(&deg[tgt[e]], 1.0f);
}
__global__ void deg3_kernel(const int* __restrict__ ei3, float* __restrict__ deg, int E3) {
    int e = blockIdx.x * blockDim.x + threadIdx.x;
    if (e >= E3) return;
    atomicAdd(&deg[ei3[e]], 1.0f);        // second half of concat: tgt = ei3[0]
    atomicAdd(&deg[ei3[E3 + e]], 1.0f);   // first half: tgt = ei3[1]
}

// ---------------------------------------------------------------------------
// Message scatter, layer 1: msum[tgt, o] += sum_k ea[e,k] * Z[src, k*64+o]
// ---------------------------------------------------------------------------
__global__ void msg1_kernel(const float* __restrict__ Z, const float* __restrict__ ea,
                            const int* __restrict__ ei, float* __restrict__ msum,
                            int E) {
    int idx = blockIdx.x * blockDim.x + threadIdx.x;
    if (idx >= E * 64) return;
    int e = idx >> 6, o = idx & 63;
    int s = ei[e], t = ei[E + e];
    const float* zp = Z + (size_t)s * 768 + o;
    const float* ep = ea + (size_t)e * 12;
    float acc = 0.0f;
    #pragma unroll
    for (int k = 0; k < 12; ++k) acc += ep[k] * zp[k * 64];
    atomicAdd(&msum[(size_t)t * 64 + o], acc);
}

// Message scatter, layer 2 over concatenated edge list (2*E3 edges, D2=128)
__global__ void msg2_kernel(const float* __restrict__ Z2, const float* __restrict__ attr3,
                            const int* __restrict__ ei3, float* __restrict__ msum,
                            int E3) {
    int idx = blockIdx.x * blockDim.x + threadIdx.x;
    int total = 2 * E3 * 128;
    if (idx >= total) return;
    int e = idx >> 7, o = idx & 127;
    int s, t;
    int eo = (e < E3) ? e : e - E3;
    if (e < E3) { s = ei3[eo]; t = ei3[E3 + eo]; }
    else        { s = ei3[E3 + eo]; t = ei3[eo]; }
    const float* zp = Z2 + (size_t)s * 1024 + o;
    const float* ap = attr3 + (size_t)eo * 8;
    float acc = 0.0f;
    #pragma unroll
    for (int k = 0; k < 8; ++k) acc += ap[k] * zp[k * 128];
    atomicAdd(&msum[(size_t)t * 128 + o], acc);
}

// m = rrelu(msum / max(deg,1) + bias)   (in place)
__global__ void finalize_m_kernel(float* __restrict__ msum, const float* __restrict__ deg,
                                  const float* __restrict__ bias, int N, int H) {
    int i = blockIdx.x * blockDim.x + threadIdx.x;
    if (i >= N * H) return;
    int n = i / H, c = i - n * H;
    float d = fmaxf(deg[n], 1.0f);
    msum[i] = rrelu_f(msum[i] / d + bias[c]);
}

// GRU pointwise: given gi[N,3H], gh[N,3H], h[N,H] -> out[N,H]
__global__ void gru_kernel(const float* __restrict__ gi, const float* __restrict__ gh,
                           const float* __restrict__ h, float* __restrict__ out,
                           int N, int H) {
    int i = blockIdx.x * blockDim.x + threadIdx.x;
    if (i >= N * H) return;
    int n = i / H, j = i - n * H;
    size_t b = (size_t)n * 3 * H;
    float ir = gi[b + j], iz = gi[b + H + j], in_ = gi[b + 2 * H + j];
    float hr = gh[b + j], hz = gh[b + H + j], hn = gh[b + 2 * H + j];
    float r = sigmoid_f(ir + hr);
    float z = sigmoid_f(iz + hz);
    float nn = tanhf(in_ + r * hn);
    float hv = h[i];
    out[i] = (1.0f - z) * nn + z * hv;
}

// ---------------------------------------------------------------------------
// Final per-edge readout (one 128-thread block per edge in E3)
// ---------------------------------------------------------------------------
__global__ void final_kernel(const float* __restrict__ out2, const float* __restrict__ attr3,
                             const int* __restrict__ ei3,
                             const float* __restrict__ lw, const float* __restrict__ lb,
                             const float* __restrict__ nm_g, const float* __restrict__ nm_be,
                             const float* __restrict__ nm_m, const float* __restrict__ nm_v,
                             float* __restrict__ res, int E3) {
    __shared__ float red[128];
    int e = blockIdx.x;
    if (e >= E3) return;
    int j = threadIdx.x;           // 0..127
    int i0 = ei3[e], i1 = ei3[E3 + e];
    float a0 = out2[(size_t)i0 * 128 + j];
    float a1 = out2[(size_t)i1 * 128 + j];
    float at[8];
    #pragma unroll
    for (int k = 0; k < 8; ++k) at[k] = attr3[(size_t)e * 8 + k];

    float f[3] = { (a0 + a1) * 0.5f, a0 * a1, (a0 - a1) * (a0 - a1) };
    float partial = 0.0f;
    #pragma unroll
    for (int g = 0; g < 3; ++g) {
        int col = g * 128 + j;
        float fb = (f[g] - nm_m[col]) * rsqrtf(nm_v[col] + BN_EPS) * nm_g[col] + nm_be[col];
        float w = 0.0f;
        #pragma unroll
        for (int k = 0; k < 8; ++k) w += at[k] * lw[(size_t)col * 8 + k];
        partial += fb * w;
    }
    red[j] = partial;
    __syncthreads();
    for (int s = 64; s > 0; s >>= 1) {
        if (j < s) red[j] += red[j + s];
        __syncthreads();
    }
    if (j == 0) {
        float bias = 0.0f;
        #pragma unroll
        for (int k = 0; k < 8; ++k) bias += at[k] * lb[k];
        res[e] = red[0] + bias;
    }
}

// ---------------------------------------------------------------------------
// Host-side orchestration
// ---------------------------------------------------------------------------
static inline int cdiv(int a, int b) { return (a + b - 1) / b; }

static void launch_gemm(const float* A, const float* W, const float* bias, float* C,
                        int M, int Nc, int K, int act, hipStream_t s) {
    int groups = (M / 32) * (Nc / 64);           // one 32x64 macro-tile per wave
    int grid = cdiv(groups, 8);                  // 8 waves per 256-thread block
    wmma_gemm_kernel<<<grid, 256, 0, s>>>(A, W, bias, C, M, Nc, K, act, groups);
}

extern "C" void kernel_launch(void* const* d_in, const int* in_sizes, int n_in,
                              void* d_out, int out_size, void* d_ws, size_t ws_size,
                              hipStream_t stream) {
    const float* x         = (const float*)d_in[0];
    const float* edge_attr = (const float*)d_in[1];
    const float* edge_attr3= (const float*)d_in[2];
    const float* nx_g = (const float*)d_in[3];
    const float* nx_be= (const float*)d_in[4];
    const float* nx_m = (const float*)d_in[5];
    const float* nx_v = (const float*)d_in[6];
    const float* ln_W = (const float*)d_in[7];
    const float* ln_b = (const float*)d_in[8];
    const float* le_W = (const float*)d_in[9];
    const float* le_b = (const float*)d_in[10];
    const float* nn1_W= (const float*)d_in[11];
    const float* c1_b = (const float*)d_in[12];
    const float* g1_wih=(const float*)d_in[13];
    const float* g1_whh=(const float*)d_in[14];
    const float* g1_bih=(const float*)d_in[15];
    const float* g1_bhh=(const float*)d_in[16];
    const float* lc_g = (const float*)d_in[17];
    const float* lc_be= (const float*)d_in[18];
    const float* lc_m = (const float*)d_in[19];
    const float* lc_v = (const float*)d_in[20];
    const float* lc_w1= (const float*)d_in[21];
    const float* lc_b1= (const float*)d_in[22];
    const float* lc_w2= (const float*)d_in[23];
    const float* lc_b2= (const float*)d_in[24];
    const float* nn2_W= (const float*)d_in[25];
    const float* c2_b = (const float*)d_in[26];
    const float* g2_wih=(const float*)d_in[27];
    const float* g2_whh=(const float*)d_in[28];
    const float* g2_bih=(const float*)d_in[29];
    const float* g2_bhh=(const float*)d_in[30];
    const float* lw_W = (const float*)d_in[31];
    const float* lb_W = (const float*)d_in[32];
    const float* nm_g = (const float*)d_in[33];
    const float* nm_be= (const float*)d_in[34];
    const float* nm_m = (const float*)d_in[35];
    const float* nm_v = (const float*)d_in[36];
    const int* edge_index  = (const int*)d_in[37];
    const int* edge_index3 = (const int*)d_in[38];

    const int N  = in_sizes[0] / 8;
    const int E  = in_sizes[1] / 19;
    const int E3 = in_sizes[2] / 8;
    float* outp = (float*)d_out;

    // -- workspace layout (floats), with aliasing of the big Z/gi/gh regions --
    float* ws = (float*)d_ws;
    size_t off = 0;
    float* xbn   = ws + off; off += (size_t)N * 8;
    float* ea    = ws + off; off += (size_t)E * 12;
    float* Wcat1 = ws + off; off += 768 * 64;
    float* Wcat2 = ws + off; off += 1024 * 128;
    float* deg   = ws + off; off += N;
    float* deg3  = ws + off; off += N;
    float* outA  = ws + off; off += (size_t)N * 64;
    float* outB  = ws + off; off += (size_t)N * 64;
    float* m1    = ws + off; off += (size_t)N * 64;   // msum1 / m1 / tmpbn reuse
    float* out2A = ws + off; off += (size_t)N * 128;
    float* out2B = ws + off; off += (size_t)N * 128;
    float* m2    = ws + off; off += (size_t)N * 128;  // msum2 / m2
    float* bigR  = ws + off; off += (size_t)N * 1024; // Z / (gi1,gh1) / Z2 / (gi2,gh2)
    float* Z   = bigR;
    float* gi1 = bigR;                       // Z dead once m1 finalized
    float* gh1 = bigR + (size_t)N * 192;
    float* Z2  = bigR;
    float* gi2 = bigR;                       // Z2 dead once m2 finalized
    float* gh2 = bigR + (size_t)N * 384;
    (void)ws_size; (void)n_in; (void)out_size;

    const int T = 256;

    // Degrees
    zero_kernel<<<cdiv(N, T), T, 0, stream>>>(deg, N);
    zero_kernel<<<cdiv(N, T), T, 0, stream>>>(deg3, N);
    deg1_kernel<<<cdiv(E, T), T, 0, stream>>>(edge_index + E, deg, E);
    deg3_kernel<<<cdiv(E3, T), T, 0, stream>>>(edge_index3, deg3, E3);

    // Embeddings + weight repacks
    bn_kernel<<<cdiv(N * 8, T), T, 0, stream>>>(x, xbn, nx_g, nx_be, nx_m, nx_v, N * 8, 8);
    launch_gemm(xbn, ln_W, ln_b, outA, N, 64, 8, /*rrelu*/1, stream);
    edge_embed_kernel<<<cdiv(E * 12, T), T, 0, stream>>>(edge_attr, le_W, le_b, ea, E);
    repack_w1_kernel<<<cdiv(768 * 64, T), T, 0, stream>>>(nn1_W, Wcat1);
    repack_w2_kernel<<<cdiv(1024 * 128, T), T, 0, stream>>>(nn2_W, Wcat2);

    // Layer 1: two NNConv+GRU rounds (h == out at loop head)
    float* cur = outA; float* nxt = outB;
    for (int r = 0; r < 2; ++r) {
        launch_gemm(cur, Wcat1, nullptr, Z, N, 768, 64, 0, stream);
        zero_kernel<<<cdiv(N * 64, T), T, 0, stream>>>(m1, N * 64);
        msg1_kernel<<<cdiv(E * 64, T), T, 0, stream>>>(Z, ea, edge_index, m1, E);
        finalize_m_kernel<<<cdiv(N * 64, T), T, 0, stream>>>(m1, deg, c1_b, N, 64);
        launch_gemm(m1,  g1_wih, g1_bih, gi1, N, 192, 64, 0, stream);
        launch_gemm(cur, g1_whh, g1_bhh, gh1, N, 192, 64, 0, stream);
        gru_kernel<<<cdiv(N * 64, T), T, 0, stream>>>(gi1, gh1, cur, nxt, N, 64);
        float* t = cur; cur = nxt; nxt = t;
    }

    // lin_covert: bn -> 64->128 rrelu -> 128->128 rrelu
    float* tmpbn = m1;
    bn_kernel<<<cdiv(N * 64, T), T, 0, stream>>>(cur, tmpbn, lc_g, lc_be, lc_m, lc_v, N * 64, 64);
    launch_gemm(tmpbn, lc_w1, lc_b1, out2B, N, 128, 64, 1, stream);
    launch_gemm(out2B, lc_w2, lc_b2, out2A, N, 128, 128, 1, stream);

    // Layer 2: two rounds over symmetrized edge list
    float* cur2 = out2A; float* nxt2 = out2B;
    for (int r = 0; r < 2; ++r) {
        launch_gemm(cur2, Wcat2, nullptr, Z2, N, 1024, 128, 0, stream);
        zero_kernel<<<cdiv(N * 128, T), T, 0, stream>>>(m2, N * 128);
        msg2_kernel<<<cdiv(2 * E3 * 128, T), T, 0, stream>>>(Z2, edge_attr3, edge_index3, m2, E3);
        finalize_m_kernel<<<cdiv(N * 128, T), T, 0, stream>>>(m2, deg3, c2_b, N, 128);
        launch_gemm(m2,   g2_wih, g2_bih, gi2, N, 384, 128, 0, stream);
        launch_gemm(cur2, g2_whh, g2_bhh, gh2, N, 384, 128, 0, stream);
        gru_kernel<<<cdiv(N * 128, T), T, 0, stream>>>(gi2, gh2, cur2, nxt2, N, 128);
        float* t = cur2; cur2 = nxt2; nxt2 = t;
    }

    // Final edge readout
    final_kernel<<<E3, 128, 0, stream>>>(cur2, edge_attr3, edge_index3, lw_W, lb_W,
                                         nm_g, nm_be, nm_m, nm_v, outp, E3);
}